// AttentiveTransformer_13580686590272
// MI455X (gfx1250) — compile-verified
//
#include <hip/hip_runtime.h>

typedef float v2f __attribute__((ext_vector_type(2)));
typedef float v8f __attribute__((ext_vector_type(8)));
typedef unsigned int uint32x4 __attribute__((ext_vector_type(4)));
typedef int          int32x8  __attribute__((ext_vector_type(8)));
typedef int          int32x4  __attribute__((ext_vector_type(4)));

#if __has_builtin(__builtin_amdgcn_tensor_load_to_lds)
#define USE_TDM 1
#else
#define USE_TDM 0
#endif

constexpr int B_    = 32768;
constexpr int INF_  = 128;
constexpr int OUTF_ = 2048;
constexpr int TM    = 16;     // rows per block
constexpr int LROW  = 2080;   // 2048 + 8 pads of 4 dwords (TDM pad: 4 dw per 256 dw)

// LDS index for (row, col) under the TDM padded layout:
// 4 DWORDs of pad inserted after every 256 DWORDs.
__device__ __forceinline__ int ldsIdx(int r, int c) {
  return r * LROW + c + ((c >> 8) << 2);
}

// fused: C = x@W (f32 WMMA), TDM-prefetched prev_mask gate (in-place in LDS),
// sparsemax per row via Michelot simplex projection.
__global__ __launch_bounds__(512, 1)
void fused_fc_gate_sparsemax(const float* __restrict__ x,
                             const float* __restrict__ pm,
                             const float* __restrict__ W,
                             float* __restrict__ out)
{
  __shared__ __align__(16) float smem[TM * LROW];   // 130 KB: pm tile, gated in place -> z tile

  const int tid  = threadIdx.x;
  const int lane = tid & 31;
  const int wv   = tid >> 5;        // wave 0..15 -> N strip [wv*128, wv*128+128)
  const int hf   = lane >> 4;       // lane half (WMMA striping)
  const int l16  = lane & 15;
  const int row0 = blockIdx.x * TM;

  // ---------- kick off the prev_mask tile DMA (overlaps the whole GEMM) ----------
#if USE_TDM
  if (wv == 0) {
    // D# group0: count=1 | lds_addr | 57-bit global byte addr | type=2
    const unsigned long long ga = (unsigned long long)(uintptr_t)(pm + (size_t)row0 * OUTF_);
    uint32x4 g0;
    g0[0] = 1u;                                             // count=1, user descriptor
    g0[1] = (unsigned int)(uintptr_t)&smem[0];              // lds_addr (low 32 bits of LDS ptr)
    g0[2] = (unsigned int)(ga & 0xFFFFFFFFu);
    g0[3] = (unsigned int)((ga >> 32) & 0x1FFFFFFu) | (2u << 30);
    // D# group1: data_size=4B, pad_enable, pad_interval=256dw, pad_amount=4dw,
    //            tensor_dim0=2048, tensor_dim1=16, tile_dim0=2048, tile_dim1=16,
    //            tensor_dim0_stride=2048
    int32x8 g1;
    g1[0] = (int)((2u << 16) | (1u << 20) | (7u << 22) | (3u << 25));
    g1[1] = (int)(2048u << 16);          // tensor_dim0[15:0] into bits[31:16]
    g1[2] = (int)(16u << 16);            // tensor_dim0 hi = 0, tensor_dim1[15:0] = 16
    g1[3] = (int)(2048u << 16);          // tensor_dim1 hi = 0, tile_dim0 = 2048
    g1[4] = 16;                          // tile_dim1 = 16, tile_dim2 = 0
    g1[5] = 2048;                        // tensor_dim0_stride lo32
    g1[6] = 0;
    g1[7] = 0;
    const int32x4 gz4 = {0, 0, 0, 0};                 // groups 2/3 unused (2D tensor)
    const int32x8 gz8 = {0, 0, 0, 0, 0, 0, 0, 0};     // extra group (6-arg toolchain form)
    __builtin_amdgcn_tensor_load_to_lds(g0, g1, gz4, gz4, gz8, 0);
  }
#else
  // Fallback: cooperative coalesced float4 copy of the pm tile into padded LDS.
  for (int i = tid * 4; i < TM * OUTF_; i += 512 * 4) {
    const int r = i >> 11;
    const int c = i & (OUTF_ - 1);
    *(float4*)&smem[ldsIdx(r, c)] = *(const float4*)&pm[(size_t)(row0 + r) * OUTF_ + c];
  }
#endif

  // ---------- GEMM: all 8 N-tiles of this wave's strip, acc kept in VGPRs ----------
  // A-frag (16x4 f32): lane L holds row M=L%16; VGPR v, half h -> K = 2*h + v
  v2f a[32];
  {
    const float* xr = x + (row0 + l16) * INF_;
    #pragma unroll
    for (int ks = 0; ks < 32; ++ks) {
      const int k0 = 4 * ks + 2 * hf;
      a[ks].x = xr[k0];
      a[ks].y = xr[k0 + 1];
    }
  }
  v8f acc[8];
  #pragma unroll 1
  for (int t = 0; t < 8; ++t) {
    const int n = wv * 128 + t * 16 + l16;
    v8f c = {0.f, 0.f, 0.f, 0.f, 0.f, 0.f, 0.f, 0.f};
    #pragma unroll
    for (int ks = 0; ks < 32; ++ks) {
      const int k0 = 4 * ks + 2 * hf;
      v2f b;
      b.x = W[k0 * OUTF_ + n];
      b.y = W[(k0 + 1) * OUTF_ + n];
      c = __builtin_amdgcn_wmma_f32_16x16x4_f32(false, a[ks], false, b,
                                                (short)0, c, false, false);
    }
    acc[t] = c;
  }

  // ---------- wait for mask DMA, then gate in place in LDS ----------
#if USE_TDM
  __builtin_amdgcn_s_wait_tensorcnt(0);   // no-op for non-issuing waves (TENSORcnt==0)
#endif
  __syncthreads();

  // C/D layout: VGPR r, lane L -> row = r + 8*(L/16), col = L%16
  #pragma unroll 1
  for (int t = 0; t < 8; ++t) {
    const int n = wv * 128 + t * 16 + l16;
    #pragma unroll
    for (int r = 0; r < 8; ++r) {
      const int lr  = r + 8 * hf;
      const int idx = ldsIdx(lr, n);
      smem[idx] = acc[t][r] * smem[idx];   // z = (x@W) * prev_mask, in place
    }
  }
  __syncthreads();

  // ---------- sparsemax: wave wv owns row wv (Michelot simplex projection) ----------
  {
    const int lr = wv;
    const float* zr = smem + lr * LROW;
    float4 z[16];
    #pragma unroll
    for (int j = 0; j < 16; ++j)
      z[j] = *(const float4*)(zr + 128 * j + 4 * lane + ((j >> 1) << 2)); // ds_load_b128

    float S = 0.f;
    #pragma unroll
    for (int j = 0; j < 16; ++j) S += (z[j].x + z[j].y) + (z[j].z + z[j].w);
    #pragma unroll
    for (int off = 16; off > 0; off >>= 1) S += __shfl_xor(S, off, 32);

    float kcnt = (float)OUTF_;
    float tau  = (S - 1.f) / kcnt;
    for (int it = 0; it < 64; ++it) {     // typically converges in ~5 iterations
      float s2 = 0.f, c2 = 0.f;
      #pragma unroll
      for (int j = 0; j < 16; ++j) {
        const float4 v = z[j];
        s2 += (v.x > tau) ? v.x : 0.f;  c2 += (v.x > tau) ? 1.f : 0.f;
        s2 += (v.y > tau) ? v.y : 0.f;  c2 += (v.y > tau) ? 1.f : 0.f;
        s2 += (v.z > tau) ? v.z : 0.f;  c2 += (v.z > tau) ? 1.f : 0.f;
        s2 += (v.w > tau) ? v.w : 0.f;  c2 += (v.w > tau) ? 1.f : 0.f;
      }
      #pragma unroll
      for (int off = 16; off > 0; off >>= 1) {
        s2 += __shfl_xor(s2, off, 32);
        c2 += __shfl_xor(c2, off, 32);
      }
      if (c2 == kcnt) break;              // active set stable -> tau exact
      kcnt = c2;
      tau  = (s2 - 1.f) / c2;
    }

    float4* op = (float4*)(out + (size_t)(row0 + lr) * OUTF_);
    #pragma unroll
    for (int j = 0; j < 16; ++j) {
      const float4 v = z[j];
      float4 o;
      o.x = fmaxf(v.x - tau, 0.f);
      o.y = fmaxf(v.y - tau, 0.f);
      o.z = fmaxf(v.z - tau, 0.f);
      o.w = fmaxf(v.w - tau, 0.f);
      op[32 * j + lane] = o;              // coalesced global_store_b128
    }
  }
}

extern "C" void kernel_launch(void* const* d_in, const int* in_sizes, int n_in,
                              void* d_out, int out_size, void* d_ws, size_t ws_size,
                              hipStream_t stream) {
  const float* x  = (const float*)d_in[0];   // [32768, 128]
  const float* pm = (const float*)d_in[1];   // [32768, 2048]
  const float* W  = (const float*)d_in[2];   // [128, 2048]
  float* out = (float*)d_out;                // [32768, 2048]

  dim3 grid(B_ / TM);                        // 2048 blocks
  dim3 block(512);                           // 16 wave32 waves
  fused_fc_gate_sparsemax<<<grid, block, 0, stream>>>(x, pm, W, out);

  (void)in_sizes; (void)n_in; (void)out_size; (void)d_ws; (void)ws_size;
}